// GraphEmb_2551210574187
// MI455X (gfx1250) — compile-verified
//
#include <hip/hip_runtime.h>
#include <hip/hip_bf16.h>

typedef __attribute__((ext_vector_type(2))) float v2f;
typedef __attribute__((ext_vector_type(8))) float v8f;

#define B_SZ 512
#define N_SZ 512
#define D_SZ 128
#define NT 4
#define ROWS_PER_TILE (N_SZ / NT)   // 128 rows per block

// ---------------------------------------------------------------------------
// Kernel 1: stream h once. Per row: gate = sigmoid(h·W_g + b_g) via wave32
// xor-shuffle reduction; accumulate a[b,d] = sum_n g*h[b,n,d] (masked) and
// S_g[b] = sum_n g. Rows >= ns[b] are skipped entirely (wave-uniform branch),
// saving their HBM traffic. Partials per (batch, tile) go to workspace.
// ---------------------------------------------------------------------------
__global__ __launch_bounds__(256)
void GraphEmb_gate_reduce(const float* __restrict__ h,
                          const int*   __restrict__ ns,
                          const float* __restrict__ W_g,
                          const float* __restrict__ b_g,
                          float* __restrict__ a_part,   // [B][NT][128]
                          float* __restrict__ gs_part)  // [B][NT]
{
    const int b    = blockIdx.y;
    const int tile = blockIdx.x;
    const int tid  = threadIdx.x;
    const int lane = tid & 31;
    const int warp = tid >> 5;          // 8 waves

    const float4 wg = ((const float4*)W_g)[lane];  // lane covers d = 4*lane..4*lane+3
    const float  bg = b_g[0];
    const int    nvalid = ns[b];

    float4 acc = make_float4(0.f, 0.f, 0.f, 0.f);
    float  gs  = 0.f;

    const float4* h4 = (const float4*)h;

    for (int r = warp; r < ROWS_PER_TILE; r += 8) {
        const int row = tile * ROWS_PER_TILE + r;
        if (row < nvalid) {   // uniform across the wave: skip masked rows' loads
            float4 hv = h4[((size_t)((size_t)b * N_SZ + row) * D_SZ) / 4 + lane];
            float d = hv.x * wg.x + hv.y * wg.y + hv.z * wg.z + hv.w * wg.w;
            #pragma unroll
            for (int off = 16; off >= 1; off >>= 1)   // wave32 reduction
                d += __shfl_xor(d, off, 32);
            float g = 1.f / (1.f + __expf(-(d + bg)));
            acc.x += g * hv.x; acc.y += g * hv.y;
            acc.z += g * hv.z; acc.w += g * hv.w;
            gs += g;                                   // same value in all lanes
        }
    }

    __shared__ float wacc[8][D_SZ];
    __shared__ float wgs[8];
    ((float4*)wacc[warp])[lane] = acc;
    if (lane == 0) wgs[warp] = gs;
    __syncthreads();

    if (tid < D_SZ) {
        float s = 0.f;
        #pragma unroll
        for (int w = 0; w < 8; ++w) s += wacc[w][tid];
        a_part[((size_t)b * NT + tile) * D_SZ + tid] = s;
    }
    if (tid == 0) {
        float s = 0.f;
        #pragma unroll
        for (int w = 0; w < 8; ++w) s += wgs[w];
        gs_part[b * NT + tile] = s;
    }
}

// ---------------------------------------------------------------------------
// Kernel 2: h_G[b,:] = a[b,:] @ W_f^T + S_g[b]*b_f, then L2-normalize.
// One block = 16 batches; 8 waves each compute one 16x16 output tile via
// 32 chained v_wmma_f32_16x16x4_f32 (K=128). Bias + norm fused in LDS.
// ---------------------------------------------------------------------------
__global__ __launch_bounds__(256)
void GraphEmb_wmma_finish(const float* __restrict__ a_part,  // [B][NT][128]
                          const float* __restrict__ gs_part, // [B][NT]
                          const float* __restrict__ W_f,     // [128 e][128 d]
                          const float* __restrict__ b_f,     // [128]
                          float* __restrict__ out)           // [B][128]
{
    const int bb   = blockIdx.x;      // 16-batch tile
    const int tid  = threadIdx.x;
    const int lane = tid & 31;
    const int warp = tid >> 5;        // e-tile 0..7

    __shared__ float a_s[16][D_SZ];
    __shared__ float out_s[16][D_SZ];
    __shared__ float sg_s[16];
    __shared__ float inv_s[16];
    __shared__ float pn[16][16];

    // sum NT partials -> a_s[m][d]; S_g per batch
    for (int i = tid; i < 16 * D_SZ; i += 256) {
        int m = i >> 7, d = i & 127;
        const float* p = a_part + ((size_t)(bb * 16 + m) * NT) * D_SZ + d;
        float s = 0.f;
        #pragma unroll
        for (int t = 0; t < NT; ++t) s += p[t * D_SZ];
        a_s[m][d] = s;
    }
    if (tid < 16) {
        const float* p = gs_part + (bb * 16 + tid) * NT;
        float s = 0.f;
        #pragma unroll
        for (int t = 0; t < NT; ++t) s += p[t];
        sg_s[tid] = s;
    }
    __syncthreads();

    // WMMA fragment indexing (ISA 16x4 f32 A layout, mirrored for B):
    //   A: M = lane&15, K = 2*(lane>>4) + {0,1}
    //   B: N = lane&15, K = 2*(lane>>4) + {0,1}
    const int mrow = lane & 15;
    const int koff = (lane >> 4) << 1;
    const int e    = warp * 16 + mrow;   // this lane's output column

    v8f c = {};
    #pragma unroll
    for (int k0 = 0; k0 < D_SZ; k0 += 4) {
        v2f af = { a_s[mrow][k0 + koff], a_s[mrow][k0 + koff + 1] };
        v2f bf = { W_f[e * D_SZ + k0 + koff], W_f[e * D_SZ + k0 + koff + 1] };
        c = __builtin_amdgcn_wmma_f32_16x16x4_f32(false, af, false, bf,
                                                  (short)0, c, false, false);
    }

    // C layout: VGPR r, lanes 0-15 -> M=r, lanes 16-31 -> M=r+8, N=lane&15
    const float bfv = b_f[e];
    #pragma unroll
    for (int r = 0; r < 8; ++r) {
        int M = r + ((lane >> 4) << 3);
        out_s[M][e] = c[r] + sg_s[M] * bfv;
    }
    __syncthreads();

    // L2 norm per batch row (128 values), then scale by 1/max(norm, eps)
    {
        int mm = tid >> 4, seg = tid & 15;
        float s = 0.f;
        #pragma unroll
        for (int j = 0; j < 8; ++j) { float v = out_s[mm][seg * 8 + j]; s += v * v; }
        pn[mm][seg] = s;
    }
    __syncthreads();
    if (tid < 16) {
        float s = 0.f;
        #pragma unroll
        for (int j = 0; j < 16; ++j) s += pn[tid][j];
        inv_s[tid] = 1.f / fmaxf(sqrtf(s), 1e-12f);
    }
    __syncthreads();
    for (int i = tid; i < 16 * D_SZ; i += 256) {
        int mm = i >> 7, ee = i & 127;
        out[(size_t)(bb * 16 + mm) * D_SZ + ee] = out_s[mm][ee] * inv_s[mm];
    }
}

extern "C" void kernel_launch(void* const* d_in, const int* in_sizes, int n_in,
                              void* d_out, int out_size, void* d_ws, size_t ws_size,
                              hipStream_t stream) {
    const float* h   = (const float*)d_in[0];
    const int*   ns  = (const int*)d_in[1];
    const float* W_f = (const float*)d_in[2];
    const float* b_f = (const float*)d_in[3];
    const float* W_g = (const float*)d_in[4];
    const float* b_g = (const float*)d_in[5];
    float* out = (float*)d_out;

    float* a_part  = (float*)d_ws;                          // B*NT*128 floats
    float* gs_part = a_part + (size_t)B_SZ * NT * D_SZ;     // B*NT floats

    GraphEmb_gate_reduce<<<dim3(NT, B_SZ), 256, 0, stream>>>(
        h, ns, W_g, b_g, a_part, gs_part);

    GraphEmb_wmma_finish<<<dim3(B_SZ / 16), 256, 0, stream>>>(
        a_part, gs_part, W_f, b_f, out);
}